// DEQ_31490700214856
// MI455X (gfx1250) — compile-verified
//
#include <hip/hip_runtime.h>
#include <math.h>

// DEQ Anderson-acceleration solver for MI455X (gfx1250, wave32).
// f(z) = tanh(C + z @ Bw^T),  C = x@Aw^T + A_b + B_b  (precomputed, loop-invariant).
// Memory-bound (~5 flop/byte) -> keep f32 end-to-end; GEMM via V_WMMA_F32_16X16X4_F32.

#define MHIST   5
#define DSTATE  128
#define LAMREG  1e-4f
#define EPS_FP  1e-3f
#define THRESH  200
#define LSTR    134   // LDS row stride (floats): even (8B-align) and bank-conflict-free

typedef float v2f __attribute__((ext_vector_type(2)));
typedef float v8f __attribute__((ext_vector_type(8)));

static __device__ __forceinline__ v8f wmma_16x16x4_f32(v2f a, v2f b, v8f c) {
  // D(16x16,f32) = A(16x4,f32) x B(4x16,f32) + C
  return __builtin_amdgcn_wmma_f32_16x16x4_f32(false, a, false, b, (short)0, c, false, false);
}

// One 16x16 output tile: A rows from xs (LDS, row-major, stride LSTR),
// B = Bw^T taken from bw LDS (Bw stored row-major [n][k], stride LSTR).
// A operand layout: lane l: M = srow + (l&15), K = 4*kk + 2*(l>>4) + {0,1}
// B operand layout: lane l: N = ncb  + (l&15), K = 4*kk + 2*(l>>4) + {0,1}
static __device__ __forceinline__ v8f gemm_tile(const float* __restrict__ xs,
                                                const float* __restrict__ bw,
                                                int srow, int ncb, int lane, v8f acc) {
  const int half2 = (lane >> 4) << 1;
  const int arow  = (srow + (lane & 15)) * LSTR;
  const int brow  = (ncb  + (lane & 15)) * LSTR;
#pragma unroll
  for (int kk = 0; kk < DSTATE / 4; ++kk) {
    const int kb = kk * 4 + half2;
    v2f a, b;
    a.x = xs[arow + kb];
    a.y = xs[arow + kb + 1];
    b.x = bw[brow + kb];
    b.y = bw[brow + kb + 1];
    acc = wmma_16x16x4_f32(a, b, acc);
  }
  return acc;
}

// ---------------------------------------------------------------- init kernels
__global__ void k_reset(float* acc4) {
  if (threadIdx.x == 0 && blockIdx.x == 0) { acc4[0] = 0.f; acc4[1] = 0.f; acc4[2] = 0.f; }
}

// C[b][n] = A_b[n] + B_b[n] + sum_j x[b][j]*Aw[n][j]   (row-major, coalesced)
__global__ void k_initC(const float* __restrict__ x, const float* __restrict__ Aw,
                        const float* __restrict__ Ab, const float* __restrict__ Bb,
                        float* __restrict__ C, int batch) {
  const int idx = blockIdx.x * blockDim.x + threadIdx.x;
  const int b = idx >> 7, n = idx & 127;
  const float* xr = x + (size_t)b * 4;
  float c = Ab[n] + Bb[n];
#pragma unroll
  for (int j = 0; j < 4; ++j) c += xr[j] * Aw[n * 4 + j];
  C[(size_t)idx] = c;
}

// f0 = tanh(C); seed history (d-major [slot][d][batch]): X0=0, X1=f0, F0=f0.
__global__ void k_init_tr(const float* __restrict__ C, float* __restrict__ X,
                          float* __restrict__ F, int batch) {
  extern __shared__ float sm[];            // DSTATE * LSTR
  const int t = threadIdx.x;
  const size_t b0 = (size_t)blockIdx.x * DSTATE;
  for (int r = 0; r < DSTATE; ++r)
    sm[r * LSTR + t] = tanhf(C[(b0 + r) * DSTATE + t]);   // coalesced row-major read
  __syncthreads();
  for (int d = 0; d < DSTATE; ++d) {                       // coalesced d-major writes
    const float v = sm[t * LSTR + d];
    const size_t o = (size_t)d * batch + b0 + t;
    X[o] = 0.f;
    X[(size_t)DSTATE * batch + o] = v;
    F[o] = v;
  }
}

// out[d][b] = tanh(C[b][d] + (zin^T)[b][:] @ Bw^T)   (zin, out in d-major layout)
__global__ void k_apply_f(const float* __restrict__ zin, float* __restrict__ outcm,
                          const float* __restrict__ C, const float* __restrict__ Bw,
                          int batch) {
  extern __shared__ float sm[];
  float* xs = sm;                    // DSTATE*LSTR: z rows for this block
  float* bw = sm + DSTATE * LSTR;    // DSTATE*LSTR: Bw row-major copy
  const int t = threadIdx.x;
  const size_t b0 = (size_t)blockIdx.x * DSTATE;
  for (int kd = 0; kd < DSTATE; ++kd)
    xs[t * LSTR + kd] = zin[(size_t)kd * batch + b0 + t];  // coalesced over b
  for (int n = 0; n < DSTATE; ++n)
    bw[n * LSTR + t] = Bw[n * DSTATE + t];
  __syncthreads();
  const int wave = t >> 5, lane = t & 31, half = lane >> 4;
#pragma unroll
  for (int sI = 0; sI < 2; ++sI) {
    const int srow = (wave + sI * 4) * 16;
    for (int nt = 0; nt < 8; ++nt) {
      const int ncb = nt * 16, ncol = ncb + (lane & 15);
      v8f acc;
#pragma unroll
      for (int r = 0; r < 8; ++r)
        acc[r] = C[(b0 + srow + half * 8 + r) * DSTATE + ncol];
      acc = gemm_tile(xs, bw, srow, ncb, lane, acc);
#pragma unroll
      for (int r = 0; r < 8; ++r)
        outcm[(size_t)ncol * batch + b0 + srow + half * 8 + r] = tanhf(acc[r]);
    }
  }
}

// -------------------------------------------------- fused Anderson iteration
__global__ void k_anderson_iter(float* __restrict__ X, float* __restrict__ F,
                                const float* __restrict__ C, const float* __restrict__ Bw,
                                float* __restrict__ zk, float* __restrict__ acc4,
                                int kiter, int batch) {
  if (acc4[2] != 0.f) return;                       // converged: whole grid no-ops
  extern __shared__ float sm[];
  float* xs  = sm;                                  // xk rows
  float* bw  = sm + DSTATE * LSTR;                  // Bw copy
  float* red = sm + 2 * DSTATE * LSTR;              // 2*DSTATE reduction scratch
  const int t = threadIdx.x;
  const size_t b0 = (size_t)blockIdx.x * DSTATE;
  const size_t b  = b0 + t;
  const int nact = kiter < MHIST ? kiter : MHIST;
  const int slot = kiter % MHIST;

  for (int n = 0; n < DSTATE; ++n) bw[n * LSTR + t] = Bw[n * DSTATE + t];

  // ---- phase 1: per-lane GGt (d-major history -> coalesced streams) ----
  float gg[MHIST][MHIST];
#pragma unroll
  for (int i = 0; i < MHIST; ++i)
#pragma unroll
    for (int j = 0; j < MHIST; ++j) gg[i][j] = 0.f;
  for (int d = 0; d < DSTATE; ++d) {
    float g[MHIST];
#pragma unroll
    for (int i = 0; i < MHIST; ++i) {
      const size_t off = (size_t)(i * DSTATE + d) * batch + b;
      const float gv = F[off] - X[off];
      g[i] = (i < nact) ? gv : 0.f;
    }
#pragma unroll
    for (int i = 0; i < MHIST; ++i)
#pragma unroll
      for (int j = i; j < MHIST; ++j) gg[i][j] += g[i] * g[j];
  }
  // bordered system reduces to (GGt+lam*I) u = 1 on active slots; alpha = u / sum(u)
  float Mm[MHIST][MHIST], u[MHIST];
#pragma unroll
  for (int i = 0; i < MHIST; ++i) {
#pragma unroll
    for (int j = 0; j < MHIST; ++j) {
      float v = (i <= j) ? gg[i][j] : gg[j][i];
      if (i == j) v += LAMREG;
      Mm[i][j] = (i < nact && j < nact) ? v : ((i == j) ? 1.f : 0.f);
    }
    u[i] = (i < nact) ? 1.f : 0.f;
  }
#pragma unroll
  for (int c = 0; c < MHIST; ++c) {                 // SPD -> no pivoting needed
    const float inv = 1.f / Mm[c][c];
#pragma unroll
    for (int r2 = c + 1; r2 < MHIST; ++r2) {
      const float fm = Mm[r2][c] * inv;
#pragma unroll
      for (int j = c; j < MHIST; ++j) Mm[r2][j] -= fm * Mm[c][j];
      u[r2] -= fm * u[c];
    }
  }
#pragma unroll
  for (int c = MHIST - 1; c >= 0; --c) {
    float s = u[c];
#pragma unroll
    for (int j = c + 1; j < MHIST; ++j) s -= Mm[c][j] * u[j];
    u[c] = s / Mm[c][c];
  }
  const float inv_s = 1.f / (u[0] + u[1] + u[2] + u[3] + u[4]);
  float alpha[MHIST];
#pragma unroll
  for (int i = 0; i < MHIST; ++i) alpha[i] = u[i] * inv_s;   // inactive -> 0

  // xk = sum_i alpha_i * F_i (beta == 1); stage to LDS, store X[slot] & zk coalesced
  for (int d = 0; d < DSTATE; ++d) {
    float s = 0.f;
#pragma unroll
    for (int i = 0; i < MHIST; ++i)
      s += alpha[i] * F[(size_t)(i * DSTATE + d) * batch + b];
    xs[t * LSTR + d] = s;
    X[(size_t)(slot * DSTATE + d) * batch + b] = s;
    zk[(size_t)d * batch + b] = s;
  }
  __syncthreads();

  // ---- phase 2: Fk = tanh(C + xk@Bw^T) via WMMA f32 16x16x4; residual sums ----
  float rn = 0.f, rd = 0.f;
  const int wave = t >> 5, lane = t & 31, half = lane >> 4;
#pragma unroll
  for (int sI = 0; sI < 2; ++sI) {
    const int srow = (wave + sI * 4) * 16;
    for (int nt = 0; nt < 8; ++nt) {
      const int ncb = nt * 16, ncol = ncb + (lane & 15);
      v8f acc;
#pragma unroll
      for (int r = 0; r < 8; ++r)
        acc[r] = C[(b0 + srow + half * 8 + r) * DSTATE + ncol];
      acc = gemm_tile(xs, bw, srow, ncb, lane, acc);
#pragma unroll
      for (int r = 0; r < 8; ++r) {
        const float fk = tanhf(acc[r]);
        const float xv = xs[(srow + half * 8 + r) * LSTR + ncol];
        const float df = fk - xv;
        rn += df * df;
        rd += fk * fk;
        F[(size_t)(slot * DSTATE + ncol) * batch + b0 + srow + half * 8 + r] = fk;
      }
    }
  }
  red[t] = rn; red[DSTATE + t] = rd;
  __syncthreads();
  for (int s2 = 64; s2 > 0; s2 >>= 1) {
    if (t < s2) { red[t] += red[t + s2]; red[DSTATE + t] += red[DSTATE + t + s2]; }
    __syncthreads();
  }
  if (t == 0) { atomicAdd(&acc4[0], red[0]); atomicAdd(&acc4[1], red[DSTATE]); }
}

__global__ void k_check(float* acc4) {
  if (threadIdx.x == 0 && blockIdx.x == 0) {
    if (acc4[2] == 0.f) {
      const float res = sqrtf(acc4[0]) / (1e-5f + sqrtf(acc4[1]));
      if (res < EPS_FP) acc4[2] = 1.f;
    }
    acc4[0] = 0.f; acc4[1] = 0.f;
  }
}

// y[b] = h_b + sum_n zstar[n][b]*h_w[n]   (zstar in d-major -> coalesced over b)
__global__ void k_final(const float* __restrict__ zs, const float* __restrict__ hw,
                        const float* __restrict__ hb, float* __restrict__ out, int batch) {
  const int bq = blockIdx.x * blockDim.x + threadIdx.x;
  float acc = hb[0];
  for (int n = 0; n < DSTATE; ++n) acc += zs[(size_t)n * batch + bq] * hw[n];
  out[bq] = acc;
}

extern "C" void kernel_launch(void* const* d_in, const int* in_sizes, int n_in,
                              void* d_out, int out_size, void* d_ws, size_t ws_size,
                              hipStream_t stream) {
  const float* x  = (const float*)d_in[0];
  const float* Aw = (const float*)d_in[1];
  const float* Ab = (const float*)d_in[2];
  const float* Bw = (const float*)d_in[3];
  const float* Bb = (const float*)d_in[4];
  const float* hw = (const float*)d_in[5];
  const float* hb = (const float*)d_in[6];
  float* out = (float*)d_out;
  const int batch = in_sizes[0] / 4;               // N_IN = 4
  const size_t nb = (size_t)batch;

  float* ws   = (float*)d_ws;                      // all f32 scratch
  float* X    = ws;                                // [5][128][batch]
  float* F    = X  + (size_t)MHIST * DSTATE * nb;  // [5][128][batch]
  float* C    = F  + (size_t)MHIST * DSTATE * nb;  // [batch][128] row-major
  float* zk   = C  + (size_t)DSTATE * nb;          // [128][batch]
  float* acc4 = zk + (size_t)DSTATE * nb;          // {res_num, res_den, done_flag}
  float* zs   = X;                                 // reuse X for z* after the loop

  const size_t ldsGemm = (size_t)(2 * DSTATE * LSTR + 2 * DSTATE) * sizeof(float);
  const size_t ldsTr   = (size_t)(DSTATE * LSTR) * sizeof(float);
  const int nBlk = batch / DSTATE;

  k_reset<<<1, 32, 0, stream>>>(acc4);
  k_initC<<<(batch * DSTATE) / 256, 256, 0, stream>>>(x, Aw, Ab, Bb, C, batch);
  k_init_tr<<<nBlk, DSTATE, ldsTr, stream>>>(C, X, F, batch);
  // F slot1 = f(f0), with f0 already in F slot0
  k_apply_f<<<nBlk, DSTATE, ldsGemm, stream>>>(F, F + (size_t)DSTATE * nb, C, Bw, batch);
  for (int k = 2; k < THRESH; ++k) {
    k_anderson_iter<<<nBlk, DSTATE, ldsGemm, stream>>>(X, F, C, Bw, zk, acc4, k, batch);
    k_check<<<1, 32, 0, stream>>>(acc4);
  }
  k_apply_f<<<nBlk, DSTATE, ldsGemm, stream>>>(zk, zs, C, Bw, batch);   // z* = f(zk)
  k_final<<<batch / 256, 256, 0, stream>>>(zs, hw, hb, out, batch);
}